// KNNGraphLearn_32512902431512
// MI455X (gfx1250) — compile-verified
//
#include <hip/hip_runtime.h>

// ---------------------------------------------------------------------------
// KNN graph (K=5) for X:(B=2, N=10000, T=64) f32 -> adjacency (10000,10000) f32
// score(i,j) = sq(j) - 2<x_i,x_j> folded into one f32 WMMA chain; streamed
// j-tiles staged via GLOBAL_LOAD_ASYNC_TO_LDS (double buffered); per-lane
// top-5 registers; half-wave shuffle merge.
// ---------------------------------------------------------------------------

#define N_NODES   10000
#define T_DIM     64
#define B_DIM     2
#define FEAT      128                 // T*B features per node
#define KPAD      132                 // [sq] + 128 feats + 3 zero pad = 33 K-chunks of 4
#define KCHUNKS   33
#define KNN       5
#define NTILES    625                 // N_NODES / 16 (exact)
#define WPB       4                   // waves per block in main kernel
#define TILE_F    (16 * KPAD)         // floats per candidate tile (2112)
#define TILE_C16  (TILE_F / 4)        // 16-byte chunks per tile (528)

typedef float v2f __attribute__((ext_vector_type(2)));
typedef float v8f __attribute__((ext_vector_type(8)));
typedef int   v4i __attribute__((vector_size(16)));   // matches builtin pointee

#if __has_builtin(__builtin_amdgcn_global_load_async_to_lds_b128)
#define USE_ASYNC_LDS 1
#endif

#if __has_builtin(__builtin_amdgcn_s_wait_asynccnt)
#define WAIT_ASYNC(n) __builtin_amdgcn_s_wait_asynccnt(n)
#else
#define WAIT_ASYNC(n) asm volatile("s_wait_asynccnt %0" ::"i"(n) : "memory")
#endif

__device__ __forceinline__ v8f wmma_f32_k4(v2f a, v2f b, v8f c) {
  // D = A(16x4 f32) * B(4x16 f32) + C(16x16 f32); wave32, VOP3P
  return __builtin_amdgcn_wmma_f32_16x16x4_f32(false, a, false, b, (short)0, c,
                                               false, false);
}

// Insert (s, j) into ascending sorted 5-list held in registers (branchless).
__device__ __forceinline__ void insert5(float s, int j,
    float& s0, float& s1, float& s2, float& s3, float& s4,
    int& i0, int& i1, int& i2, int& i3, int& i4) {
  bool c = s < s4;
  s4 = c ? s : s4;  i4 = c ? j : i4;
#define SORT2(sa, sb, ia, ib)                                              \
  { bool cc = (sb) < (sa);                                                 \
    float tf = cc ? (sb) : (sa); (sb) = cc ? (sa) : (sb); (sa) = tf;       \
    int ti = cc ? (ib) : (ia);  (ib) = cc ? (ia) : (ib); (ia) = ti; }
  SORT2(s3, s4, i3, i4);
  SORT2(s2, s3, i2, i3);
  SORT2(s1, s2, i1, i2);
  SORT2(s0, s1, i0, i1);
#undef SORT2
}

// ------------------------- kernel 1: zero output ---------------------------
__global__ void zero_out_kernel(float4* __restrict__ out, int n4) {
  int stride = gridDim.x * blockDim.x;
  for (int i = blockIdx.x * blockDim.x + threadIdx.x; i < n4; i += stride) {
    out[i] = make_float4(0.f, 0.f, 0.f, 0.f);
  }
}

// ------------------------- kernel 2: prep features -------------------------
// Acand[n] = [ sq(n), x(n,0..127), 0,0,0 ]        (candidate rows, WMMA A)
// Bqry[n]  = [ 1.0,  -2*x(n,0..127), 0,0,0 ]      (query rows,     WMMA B)
// Feature order b*64+t (dot products are order invariant vs. reference t*2+b).
__global__ __launch_bounds__(FEAT) void prep_kernel(
    const float* __restrict__ X, float* __restrict__ Ac, float* __restrict__ Bq) {
  const int n = blockIdx.x;           // one block of 128 threads per node
  const int f = threadIdx.x;          // feature id 0..127
  __shared__ float red[FEAT];

  const int b = f >> 6, t = f & 63;
  float x = X[(size_t)b * (N_NODES * T_DIM) + (size_t)n * T_DIM + t];

  Ac[(size_t)n * KPAD + 1 + f] = x;
  Bq[(size_t)n * KPAD + 1 + f] = -2.0f * x;
  red[f] = x * x;
  __syncthreads();
  for (int s = FEAT / 2; s > 0; s >>= 1) {
    if (f < s) red[f] += red[f + s];
    __syncthreads();
  }
  if (f == 0) { Ac[(size_t)n * KPAD] = red[0]; Bq[(size_t)n * KPAD] = 1.0f; }
  if (f < 3)  { Ac[(size_t)n * KPAD + 129 + f] = 0.0f;
                Bq[(size_t)n * KPAD + 129 + f] = 0.0f; }
}

// ------------------------- kernel 3: WMMA + streaming top-5 ----------------
// Wave w handles query tile itile = blockIdx*WPB + w (16 queries, WMMA N dim).
// Loop over 625 candidate tiles (WMMA M dim); 4 waves share each candidate
// tile, staged double-buffered in LDS via GLOBAL_LOAD_ASYNC_TO_LDS_B128.
// Lane layout (f32 16x16x4):
//   A frag: lane holds row (lane&15), K = kk*4 + (lane<16 ? {0,1} : {2,3})
//   B frag: same indexing against query rows -> loop invariant, in registers
//   C/D:    lane L, vgpr v -> M = (L<16 ? v : v+8), N = L&15
__global__ __launch_bounds__(32 * WPB) void knn_topk_kernel(
    const float* __restrict__ Ac, const float* __restrict__ Bq,
    int* __restrict__ topk) {
  __shared__ __align__(16) float tile[2][TILE_F];   // 2 x 8448 B candidate tiles

  const int tid   = threadIdx.x;
  const int lane  = tid & 31;
  const int wave  = tid >> 5;
  const int itile = blockIdx.x * WPB + wave;
  const bool active = (itile < NTILES);

  const int qcol  = lane & 15;                       // N index in tile
  const int q     = itile * 16 + qcol;               // this lane's query id
  const int khalf = (lane < 16) ? 0 : 2;             // K offset by half-wave
  const int mbase = (lane < 16) ? 0 : 8;             // M base for C/D vgprs

  // Stage one 8448B candidate tile into LDS buffer `buf`.
  // Each wave issues exactly 5 b128 async-loads (uniform ASYNCcnt bookkeeping):
  // chunk index clamped for the tail round -> duplicate write of same data.
  auto stage_tile = [&](int jt, int buf) {
    const char* src = (const char*)(Ac + (size_t)jt * TILE_F);
    char* dst = (char*)&tile[buf][0];
#ifdef USE_ASYNC_LDS
#pragma unroll
    for (int r = 0; r < 5; ++r) {
      int c = tid + r * (32 * WPB);
      c = (c < TILE_C16) ? c : (c - TILE_C16);
      __builtin_amdgcn_global_load_async_to_lds_b128(
          (__attribute__((address_space(1))) v4i*)(src + (size_t)c * 16),
          (__attribute__((address_space(3))) v4i*)(dst + c * 16), 0, 0);
    }
#else
    for (int c = tid; c < TILE_C16; c += 32 * WPB)
      ((float4*)dst)[c] = ((const float4*)src)[c];
#endif
  };

  // Load loop-invariant query B-fragments (33 chunks -> 66 VGPRs).
  v2f bq[KCHUNKS];
  {
    const float* brow = Bq + (size_t)(active ? q : 0) * KPAD + khalf;
#pragma unroll
    for (int kk = 0; kk < KCHUNKS; ++kk)
      bq[kk] = *(const v2f*)(brow + kk * 4);
  }

  // Per-lane running top-5 (ascending scores).
  float s0, s1, s2, s3, s4;
  int   i0, i1, i2, i3, i4;
  s0 = s1 = s2 = s3 = s4 = __builtin_inff();
  i0 = i1 = i2 = i3 = i4 = 0;

  stage_tile(0, 0);                                  // prologue: tile 0 in flight

  for (int jt = 0; jt < NTILES; ++jt) {
    const int cur = jt & 1;
    if (jt + 1 < NTILES) {
      stage_tile(jt + 1, cur ^ 1);                   // overlap next tile's DMA
      if (jt + 2 < NTILES)                           // warm L2 two tiles ahead
        __builtin_prefetch(Ac + (size_t)(jt + 2) * TILE_F + tid * 16, 0, 1);
#ifdef USE_ASYNC_LDS
      WAIT_ASYNC(5);                                 // current tile's 5 done
#endif
    } else {
#ifdef USE_ASYNC_LDS
      WAIT_ASYNC(0);
#endif
    }
    __syncthreads();                                 // tile[cur] visible to all

    // 33 chunk WMMAs, two accumulators to break the RAW chain.
    v8f acc0 = {}, acc1 = {};
    const float* arow = &tile[cur][0] + qcol * KPAD + khalf;
#pragma unroll
    for (int kk = 0; kk < KCHUNKS; ++kk) {
      v2f a = *(const v2f*)(arow + kk * 4);          // ds_load_b64
      if (kk & 1) acc1 = wmma_f32_k4(a, bq[kk], acc1);
      else        acc0 = wmma_f32_k4(a, bq[kk], acc0);
    }

    if (active) {
#pragma unroll
      for (int v = 0; v < 8; ++v) {
        int   j = jt * 16 + mbase + v;
        float s = acc0[v] + acc1[v];                 // sq(j) - 2<x_i,x_j>
        s = (j == q) ? __builtin_inff() : s;         // exclude self
        insert5(s, j, s0, s1, s2, s3, s4, i0, i1, i2, i3, i4);
      }
    }
    __syncthreads();                                 // before buf[cur^1->cur] reuse
  }

  // Lanes L and L+16 cover disjoint candidate halves for the same query:
  // exchange and merge (shuffles are wave-uniform; 'active' is too).
  {
    float ps; int pj;
    ps = __shfl_xor(s0, 16, 32); pj = __shfl_xor(i0, 16, 32);
    insert5(ps, pj, s0, s1, s2, s3, s4, i0, i1, i2, i3, i4);
    ps = __shfl_xor(s1, 16, 32); pj = __shfl_xor(i1, 16, 32);
    insert5(ps, pj, s0, s1, s2, s3, s4, i0, i1, i2, i3, i4);
    ps = __shfl_xor(s2, 16, 32); pj = __shfl_xor(i2, 16, 32);
    insert5(ps, pj, s0, s1, s2, s3, s4, i0, i1, i2, i3, i4);
    ps = __shfl_xor(s3, 16, 32); pj = __shfl_xor(i3, 16, 32);
    insert5(ps, pj, s0, s1, s2, s3, s4, i0, i1, i2, i3, i4);
    ps = __shfl_xor(s4, 16, 32); pj = __shfl_xor(i4, 16, 32);
    insert5(ps, pj, s0, s1, s2, s3, s4, i0, i1, i2, i3, i4);
  }

  if (active && lane < 16) {
    int* t = topk + (size_t)q * KNN;
    t[0] = i0; t[1] = i1; t[2] = i2; t[3] = i3; t[4] = i4;
  }
}

// ------------------------- kernel 4: scatter adjacency ---------------------
__global__ void scatter_kernel(const int* __restrict__ topk,
                               float* __restrict__ out) {
  int t = blockIdx.x * blockDim.x + threadIdx.x;
  if (t >= N_NODES * KNN) return;
  int i = t / KNN;
  int j = topk[t];
  out[(size_t)i * N_NODES + j] = 1.0f;   // A[i,j] = 1
  out[(size_t)j * N_NODES + i] = 1.0f;   // A[j,i] = 1 (symmetric OR)
}

// ---------------------------------------------------------------------------
extern "C" void kernel_launch(void* const* d_in, const int* in_sizes, int n_in,
                              void* d_out, int out_size, void* d_ws, size_t ws_size,
                              hipStream_t stream) {
  (void)in_sizes; (void)n_in; (void)ws_size; (void)out_size;
  const float* X = (const float*)d_in[0];
  float* out = (float*)d_out;

  // Workspace layout (needs ~10.8 MB):
  float* Ac   = (float*)d_ws;                          // 10000*132 f32
  float* Bq   = Ac + (size_t)N_NODES * KPAD;           // 10000*132 f32
  int*   topk = (int*)(Bq + (size_t)N_NODES * KPAD);   // 10000*5 i32

  // 1) zero 400 MB adjacency
  zero_out_kernel<<<4096, 256, 0, stream>>>((float4*)out,
                                            (N_NODES * N_NODES) / 4);
  // 2) prep padded feature buffers (+sq fold)
  prep_kernel<<<N_NODES, FEAT, 0, stream>>>(X, Ac, Bq);
  // 3) WMMA gram + streaming top-5
  knn_topk_kernel<<<(NTILES + WPB - 1) / WPB, 32 * WPB, 0, stream>>>(Ac, Bq, topk);
  // 4) scatter symmetric ones
  scatter_kernel<<<(N_NODES * KNN + 255) / 256, 256, 0, stream>>>(topk, out);
}